// EqualizedOddsLoss_19464791786316
// MI455X (gfx1250) — compile-verified
//
#include <hip/hip_runtime.h>
#include <hip/hip_bf16.h>

#define N_SAMPLES   1048576
#define NUM_CLASSES 100
#define NUM_GROUPS  4
#define NBLK        2048        // main-pass blocks; also # of CE partials
#define GC          (NUM_GROUPS * NUM_CLASSES)   // 400

typedef __attribute__((ext_vector_type(4))) float v4f;
typedef __attribute__((ext_vector_type(2))) float v2f;
typedef __attribute__((ext_vector_type(8))) float v8f;

// ---------------------------------------------------------------------------
// Kernel 0: zero the workspace accumulators (3*400 u32 hist + NBLK f32 CE).
// ---------------------------------------------------------------------------
__global__ void eo_zero_ws(unsigned* __restrict__ ws) {
    int i = blockIdx.x * blockDim.x + threadIdx.x;
    if (i < 3 * GC + NBLK) ws[i] = 0u;
}

// ---------------------------------------------------------------------------
// Kernel 1: streaming pass. One wave32 per sample row (100 f32 = 25 lanes x
// float4, perfectly coalesced, non-temporal). Shuffle-reduce max/argmax and
// sum(exp), LDS u32 histograms, per-block CE partial.
// ---------------------------------------------------------------------------
__global__ __launch_bounds__(256) void eo_main(
    const float* __restrict__ logits,
    const int*   __restrict__ targets,
    const int*   __restrict__ groups,
    unsigned* __restrict__ gTP, unsigned* __restrict__ gP,
    unsigned* __restrict__ gPC, float* __restrict__ ceBlk)
{
    __shared__ unsigned sTP[GC], sP[GC], sPC[GC];
    __shared__ float    ceW[8];

    const int tid  = threadIdx.x;
    const int lane = tid & 31;
    const int wid  = tid >> 5;

    for (int i = tid; i < GC; i += 256) { sTP[i] = 0u; sP[i] = 0u; sPC[i] = 0u; }
    __syncthreads();

    const int gwid = blockIdx.x * 8 + wid;
    const int nw   = gridDim.x * 8;

    float ce_acc = 0.f;

    for (int s = gwid; s < N_SAMPLES; s += nw) {
        const v4f* row = (const v4f*)(logits + (size_t)s * NUM_CLASSES);

        float m  = -3.402823466e+38f;
        int   mi = 0x3FFFFFFF;
        v4f   v  = {};
        if (lane < 25) {
            v = __builtin_nontemporal_load(row + lane);    // b128 TH=NT
            m = v.x; mi = lane * 4;
            if (v.y > m) { m = v.y; mi = lane * 4 + 1; }
            if (v.z > m) { m = v.z; mi = lane * 4 + 2; }
            if (v.w > m) { m = v.w; mi = lane * 4 + 3; }
            // prefetch next row this lane will touch
            if (s + nw < N_SAMPLES)
                __builtin_prefetch((const void*)(logits + (size_t)(s + nw) * NUM_CLASSES + lane * 4), 0, 1);
        }
        // wave32 max + first-occurrence argmax
        #pragma unroll
        for (int off = 16; off; off >>= 1) {
            float om = __shfl_xor(m, off, 32);
            int   oi = __shfl_xor(mi, off, 32);
            if (om > m || (om == m && oi < mi)) { m = om; mi = oi; }
        }
        // wave32 sum(exp(x - max))
        float es = 0.f;
        if (lane < 25)
            es = __expf(v.x - m) + __expf(v.y - m) + __expf(v.z - m) + __expf(v.w - m);
        #pragma unroll
        for (int off = 16; off; off >>= 1) es += __shfl_xor(es, off, 32);

        const int t = targets[s];
        const int comp = t & 3;
        float xt_local = (comp == 0) ? v.x : (comp == 1) ? v.y : (comp == 2) ? v.z : v.w;
        float xt = __shfl(xt_local, t >> 2, 32);

        ce_acc += (m - xt) + __logf(es);   // -log_softmax[target]

        if (lane == 0) {
            const int g = groups[s];
            atomicAdd(&sP [g * NUM_CLASSES + t ], 1u);
            atomicAdd(&sPC[g * NUM_CLASSES + mi], 1u);
            if (mi == t) atomicAdd(&sTP[g * NUM_CLASSES + t], 1u);
        }
    }

    if (lane == 0) ceW[wid] = ce_acc;
    __syncthreads();

    for (int i = tid; i < GC; i += 256) {
        if (sTP[i]) atomicAdd(&gTP[i], sTP[i]);
        if (sP [i]) atomicAdd(&gP [i], sP [i]);
        if (sPC[i]) atomicAdd(&gPC[i], sPC[i]);
    }
    if (tid == 0) {
        float c = 0.f;
        #pragma unroll
        for (int w = 0; w < 8; ++w) c += ceW[w];   // fixed order: deterministic
        ceBlk[blockIdx.x] = c;
    }
}

// ---------------------------------------------------------------------------
// Kernel 2: finalize. Deterministic CE reduction (double), TPR/FPR from the
// exact integer histograms, then the 800 squared deviations are reduced on
// the matrix unit: 13 x V_WMMA_F32_16X16X4_F32 accumulations vs an all-ones
// B matrix (D[i][j] = sum_k A[i][k]); column 0 of D holds the row sums.
// ---------------------------------------------------------------------------
__global__ __launch_bounds__(512) void eo_final(
    const unsigned* __restrict__ gTP, const unsigned* __restrict__ gP,
    const unsigned* __restrict__ gPC, const float* __restrict__ ceBlk,
    float* __restrict__ out)
{
    __shared__ double red[512];
    __shared__ float  grpL[NUM_GROUPS];
    __shared__ float  tprL[GC], fprL[GC];
    __shared__ float  stage[832];          // 13 * 64, zero-padded past 800
    __shared__ float  totalL;

    const int tid = threadIdx.x;

    // CE partial sums -> double tree reduce (fixed partition => deterministic)
    double cs = 0.0;
    for (int i = tid; i < NBLK; i += 512) cs += (double)ceBlk[i];
    red[tid] = cs;
    __syncthreads();
    for (int s = 256; s > 0; s >>= 1) {
        if (tid < s) red[tid] += red[tid + s];
        __syncthreads();
    }

    // group totals = sum_c P[g,c]
    if (tid < NUM_GROUPS) {
        unsigned acc = 0u;
        for (int c = 0; c < NUM_CLASSES; ++c) acc += gP[tid * NUM_CLASSES + c];
        grpL[tid] = (float)acc;
    }
    for (int i = tid; i < 832; i += 512) stage[i] = 0.f;
    __syncthreads();

    // tpr/fpr, laid out [c*4 + g] so the 4 group values per class are adjacent
    if (tid < GC) {
        const int g = tid & 3, c = tid >> 2;
        const float TPv = (float)gTP[g * NUM_CLASSES + c];
        const float Pv  = (float)gP [g * NUM_CLASSES + c];
        const float PCv = (float)gPC[g * NUM_CLASSES + c];
        const float FPv  = PCv - TPv;
        const float NEGv = grpL[g] - Pv;
        tprL[tid] = (Pv   > 0.f) ? TPv / fmaxf(Pv,   1.f) : 0.f;
        fprL[tid] = (NEGv > 0.f) ? FPv / fmaxf(NEGv, 1.f) : 0.f;
    }
    __syncthreads();

    if (tid < GC) {
        const int b = (tid >> 2) * 4;
        const float mT = 0.25f * (tprL[b] + tprL[b+1] + tprL[b+2] + tprL[b+3]);
        const float mF = 0.25f * (fprL[b] + fprL[b+1] + fprL[b+2] + fprL[b+3]);
        const float dT = tprL[tid] - mT;
        const float dF = fprL[tid] - mF;
        stage[tid]       = dT * dT;
        stage[GC + tid]  = dF * dF;
    }
    __syncthreads();

    // WMMA reduction: wave 0, all 32 lanes active (EXEC all-ones in branch).
    if (tid < 32) {
        const int lane = tid;
        const int mrow = lane & 15;
        const int kb   = (lane & 16) ? 2 : 0;   // 16x4 f32 A layout
        v8f c = {};
        v2f bm; bm.x = 1.f; bm.y = 1.f;         // all-ones B (4x16)
        #pragma unroll
        for (int chunk = 0; chunk < 13; ++chunk) {
            v2f a;
            const int base = chunk * 64 + mrow * 4 + kb;
            a.x = stage[base];
            a.y = stage[base + 1];
            c = __builtin_amdgcn_wmma_f32_16x16x4_f32(
                    false, a, false, bm, (short)0, c, false, false);
        }
        // column 0: lane 0 holds M=0..7, lane 16 holds M=8..15
        float srow = c[0] + c[1] + c[2] + c[3] + c[4] + c[5] + c[6] + c[7];
        srow += __shfl_xor(srow, 16, 32);
        if (lane == 0) totalL = srow;
    }
    __syncthreads();

    if (tid == 0) {
        const float ce  = (float)(red[0] / (double)N_SAMPLES);
        const float pen = totalL * (1.0f / (float)GC);   // (sumT + sumF)/400
        out[0] = ce + 0.1f * pen;
    }
}

// ---------------------------------------------------------------------------
extern "C" void kernel_launch(void* const* d_in, const int* in_sizes, int n_in,
                              void* d_out, int out_size, void* d_ws, size_t ws_size,
                              hipStream_t stream) {
    const float* logits  = (const float*)d_in[0];
    const int*   targets = (const int*)  d_in[1];
    const int*   groups  = (const int*)  d_in[2];

    unsigned* ws  = (unsigned*)d_ws;
    unsigned* gTP = ws;
    unsigned* gP  = ws + GC;
    unsigned* gPC = ws + 2 * GC;
    float*    ceBlk = (float*)(ws + 3 * GC);

    const int zTotal = 3 * GC + NBLK;
    eo_zero_ws<<<(zTotal + 255) / 256, 256, 0, stream>>>(ws);
    eo_main<<<NBLK, 256, 0, stream>>>(logits, targets, groups, gTP, gP, gPC, ceBlk);
    eo_final<<<1, 512, 0, stream>>>(gTP, gP, gPC, ceBlk, (float*)d_out);
}